// Decoder_19387482374827
// MI455X (gfx1250) — compile-verified
//
#include <hip/hip_runtime.h>
#include <hip/hip_bf16.h>
#include <math.h>

// Sizes from the reference
#define BB   512
#define TM1  127
#define ENC  256
#define DEC  256

typedef __attribute__((ext_vector_type(2))) float v2f;
typedef __attribute__((ext_vector_type(8))) float v8f;

// CDNA5 fp32 WMMA: D(16x16,f32) = A(16x4,f32) * B(4x16,f32) + C
static __device__ __forceinline__ v8f wmma_f32_k4(v2f a, v2f b, v8f c) {
  return __builtin_amdgcn_wmma_f32_16x16x4_f32(false, a, false, b, (short)0, c,
                                               false, false);
}

static __device__ __forceinline__ float sigmoidf_(float x) {
  return 1.0f / (1.0f + __expf(-x));
}

// gfx1250 has V_TANH_F32 (TRANS op); use it for the attention scores.
#if __has_builtin(__builtin_amdgcn_tanh_f32)
static __device__ __forceinline__ float fast_tanh(float x) {
  return __builtin_amdgcn_tanh_f32(x);
}
#elif __has_builtin(__builtin_amdgcn_tanhf)
static __device__ __forceinline__ float fast_tanh(float x) {
  return __builtin_amdgcn_tanhf(x);
}
#else
static __device__ __forceinline__ float fast_tanh(float x) { return tanhf(x); }
#endif

// ---------------------------------------------------------------------------
// Prep: transpose W_attn1 (256x768) into K-pair-interleaved layouts so every
// WMMA B fetch is ONE aligned 8-byte load:
//   element (k, n) -> buf[ ((k>>1)*N + n)*2 + (k&1) ]
// Wt_hc: K=512,N=256 (h|c cols). Wt_e: K=256,N=256 (enc cols).
// ---------------------------------------------------------------------------
__global__ __launch_bounds__(256) void k_prep_attn_w(const float* __restrict__ W1,
                                                     float* __restrict__ Wt_hc,
                                                     float* __restrict__ Wt_e) {
  int idx = blockIdx.x * 256 + threadIdx.x;   // 0 .. 768*256-1
  int j = idx & 255;
  int k = idx >> 8;                           // 0..767
  float v = W1[(size_t)j * 768 + k];
  if (k < 512) {
    Wt_hc[(((size_t)(k >> 1)) * 256 + j) * 2 + (k & 1)] = v;
  } else {
    int km = k - 512;
    Wt_e[(((size_t)(km >> 1)) * 256 + j) * 2 + (km & 1)] = v;
  }
}

// Prep: W_hh^T K-pair-interleaved (K=256,N=1024) + combined gate bias.
__global__ __launch_bounds__(256) void k_prep_hh(const float* __restrict__ Whh,
                                                 const float* __restrict__ bih,
                                                 const float* __restrict__ bhh,
                                                 float* __restrict__ Wt_hh,
                                                 float* __restrict__ biasg) {
  int idx = blockIdx.x * 256 + threadIdx.x;   // 0 .. 256*1024-1
  int n = idx & 1023;
  int k = idx >> 10;                          // 0..255
  Wt_hh[(((size_t)(k >> 1)) * 1024 + n) * 2 + (k & 1)] = Whh[(size_t)n * 256 + k];
  if (idx < 4 * DEC) biasg[idx] = bih[idx] + bhh[idx];
}

__global__ __launch_bounds__(256) void k_zero(float* __restrict__ p, int n) {
  int idx = blockIdx.x * 256 + threadIdx.x;
  if (idx < n) p[idx] = 0.0f;
}

// ---------------------------------------------------------------------------
// E_proj[m, n] = enc[m, :] . W1e^T[:, n] + b_attn1[n],  m in [0, B*127)
// One WG = 16 rows of M; 8 waves x 2 N-tiles; K=256 via 64 chained f32 WMMAs.
// ---------------------------------------------------------------------------
__global__ __launch_bounds__(256) void k_eproj(const float* __restrict__ enc,
                                               const float* __restrict__ Wt_e,
                                               const float* __restrict__ b1,
                                               float* __restrict__ Eproj) {
  __shared__ float aT[16 * 257];              // padded: conflict-free A loads
  const int tid  = threadIdx.x;
  const int lane = tid & 31, wave = tid >> 5;
  const size_t m0 = (size_t)blockIdx.x * 16;
  const v2f* __restrict__ Wp = (const v2f*)Wt_e;   // [128][256] K-pairs

  for (int idx = tid; idx < 16 * 64; idx += 256) {
    int r = idx >> 6, k4 = (idx & 63) << 2;
    float4 v = *(const float4*)&enc[(m0 + r) * 256 + k4];
    aT[r * 257 + k4 + 0] = v.x;
    aT[r * 257 + k4 + 1] = v.y;
    aT[r * 257 + k4 + 2] = v.z;
    aT[r * 257 + k4 + 3] = v.w;
  }
  __syncthreads();

  const int hl = lane >> 4;                   // half-wave: 0/1
  const int ml = lane & 15;
  v8f acc[2];
#pragma unroll
  for (int q = 0; q < 2; ++q) {
    int n = (wave * 2 + q) * 16 + ml;
    float bv = b1[n];
#pragma unroll
    for (int i = 0; i < 8; ++i) acc[q][i] = bv;
  }
  for (int k0 = 0; k0 < 256; k0 += 4) {
    int ka = k0 + (hl << 1);
    v2f a;
    a.x = aT[ml * 257 + ka];
    a.y = aT[ml * 257 + ka + 1];
#pragma unroll
    for (int q = 0; q < 2; ++q) {
      int n = (wave * 2 + q) * 16 + ml;
      v2f bv = Wp[(size_t)(ka >> 1) * 256 + n];        // one b64 load
      acc[q] = wmma_f32_k4(a, bv, acc[q]);
    }
  }
#pragma unroll
  for (int q = 0; q < 2; ++q) {
    int n = (wave * 2 + q) * 16 + ml;
#pragma unroll
    for (int i = 0; i < 8; ++i) {
      int M = (hl << 3) + i;
      Eproj[(m0 + M) * 256 + n] = acc[q][i];
    }
  }
}

// ---------------------------------------------------------------------------
// Per-step attention. One WG per batch row b.
// score[t] = sum_j w2[j]*tanh(u[b,j] + Eproj[b,t,j])   (b_attn2 dropped:
// softmax-invariant). Softmax -> attn row written straight into d_out weight,
// context, and scalar y_tilde.
// ---------------------------------------------------------------------------
__global__ __launch_bounds__(256) void k_attn(const float* __restrict__ Eproj,
                                              const float* __restrict__ enc,
                                              const float* __restrict__ u,
                                              const float* __restrict__ W2,
                                              const float* __restrict__ Wfc,
                                              const float* __restrict__ bfc,
                                              const float* __restrict__ yhist,
                                              float* __restrict__ ctx,
                                              float* __restrict__ ytl,
                                              float* __restrict__ wout,
                                              int s) {
  __shared__ __align__(16) float u_s[256];
  __shared__ __align__(16) float w2_s[256];
  __shared__ __align__(16) float ctxred[4 * 256];
  __shared__ float sc[128], red[8];
  const int tid  = threadIdx.x;
  const int lane = tid & 31, wave = tid >> 5;
  const int b = blockIdx.x;

  u_s[tid]  = u[(size_t)b * 256 + tid];
  w2_s[tid] = W2[tid];
  __syncthreads();

  // Loop-invariant per-lane slices (registers)
  const float4* u4 = (const float4*)u_s;
  const float4* w4 = (const float4*)w2_s;
  const float4 ua = u4[lane], ub = u4[lane + 32];
  const float4 wa = w4[lane], wb = w4[lane + 32];

  const float4* Eb4 = (const float4*)(Eproj + (size_t)b * TM1 * 256); // [127][64]
  for (int t0 = 0; t0 < 128; t0 += 8) {       // wave w handles t = t0 + w
    int t = t0 + wave;
    float acc = 0.0f;
    if (t < TM1) {
      float4 r0 = Eb4[(size_t)t * 64 + lane];
      float4 r1 = Eb4[(size_t)t * 64 + lane + 32];
      acc  = wa.x * fast_tanh(ua.x + r0.x);
      acc += wa.y * fast_tanh(ua.y + r0.y);
      acc += wa.z * fast_tanh(ua.z + r0.z);
      acc += wa.w * fast_tanh(ua.w + r0.w);
      acc += wb.x * fast_tanh(ub.x + r1.x);
      acc += wb.y * fast_tanh(ub.y + r1.y);
      acc += wb.z * fast_tanh(ub.z + r1.z);
      acc += wb.w * fast_tanh(ub.w + r1.w);
    }
#pragma unroll
    for (int off = 16; off; off >>= 1) acc += __shfl_xor(acc, off, 32);
    if (t < TM1 && lane == 0) sc[t] = acc;
  }
  if (tid == 0) sc[127] = -3.0e38f;           // pad for reductions
  __syncthreads();

  if (wave == 0) {                            // softmax over 127 (wave 0 only)
    float s0 = sc[lane], s1 = sc[lane + 32], s2 = sc[lane + 64], s3 = sc[lane + 96];
    float m = fmaxf(fmaxf(s0, s1), fmaxf(s2, s3));
#pragma unroll
    for (int off = 16; off; off >>= 1) m = fmaxf(m, __shfl_xor(m, off, 32));
    float e0 = __expf(s0 - m), e1 = __expf(s1 - m);
    float e2 = __expf(s2 - m), e3 = __expf(s3 - m);
    float t = e0 + e1 + e2 + e3;
#pragma unroll
    for (int off = 16; off; off >>= 1) t += __shfl_xor(t, off, 32);
    float inv = 1.0f / t;
    sc[lane] = e0 * inv; sc[lane + 32] = e1 * inv;
    sc[lane + 64] = e2 * inv; sc[lane + 96] = e3 * inv;
  }
  __syncthreads();

  if (tid < TM1) wout[((size_t)b * TM1 + s) * TM1 + tid] = sc[tid];

  // context: thread = (t-group tg, column-quad jq); float4 partials + LDS add
  const int jq = tid & 63, tg = tid >> 6;
  const float4* eb4 = (const float4*)(enc + (size_t)b * TM1 * 256);  // [127][64]
  float4 cp = make_float4(0.f, 0.f, 0.f, 0.f);
  for (int t = tg; t < TM1; t += 4) {
    float a = sc[t];
    float4 e = eb4[(size_t)t * 64 + jq];
    cp.x += a * e.x; cp.y += a * e.y; cp.z += a * e.z; cp.w += a * e.w;
  }
  ((float4*)ctxred)[tg * 64 + jq] = cp;
  __syncthreads();
  float ctxv = ctxred[tid] + ctxred[256 + tid] + ctxred[512 + tid] +
               ctxred[768 + tid];
  ctx[(size_t)b * 256 + tid] = ctxv;

  float v = Wfc[tid] * ctxv;                  // y_tilde = Wfc.[ctx, y_t] + b
#pragma unroll
  for (int off = 16; off; off >>= 1) v += __shfl_xor(v, off, 32);
  if (lane == 0) red[wave] = v;
  __syncthreads();
  if (tid == 0) {
    float tot = 0.0f;
#pragma unroll
    for (int w = 0; w < 8; ++w) tot += red[w];
    ytl[b] = tot + Wfc[256] * yhist[(size_t)b * TM1 + s] + bfc[0];
  }
}

// ---------------------------------------------------------------------------
// Per-step LSTM + next-step h/c projection. One WG = 16 batch rows, 8 waves.
//  gates[16,1024] = ytilde x W_ih + h.Whh^T + (b_ih+b_hh)   (f32 WMMA)
//  pointwise LSTM update (h,c), then u_next[16,256] = [h,c].W1hc^T (f32 WMMA)
// Wave w owns columns j in [32w, 32w+32) of ALL four gates -> no cross-wave
// gate exchange; h/c tile lives in padded LDS (stride 521, conflict-free).
// ---------------------------------------------------------------------------
#define HCSTR 521
__global__ __launch_bounds__(256) void k_lstm(float* __restrict__ h,
                                              float* __restrict__ c,
                                              float* __restrict__ u,
                                              const float* __restrict__ ytl,
                                              const float* __restrict__ Wih,
                                              const float* __restrict__ biasg,
                                              const float* __restrict__ Wt_hh,
                                              const float* __restrict__ Wt_hc) {
  __shared__ float hc[16 * HCSTR];            // [r][0:256)=h, [r][256:512)=c
  __shared__ float yt_s[16];
  const int tid  = threadIdx.x;
  const int lane = tid & 31, wave = tid >> 5;
  const int wg = blockIdx.x;
  const v2f* __restrict__ Whh2 = (const v2f*)Wt_hh;   // [128][1024] K-pairs
  const v2f* __restrict__ Whc2 = (const v2f*)Wt_hc;   // [256][256]  K-pairs

  for (int idx = tid; idx < 16 * 64; idx += 256) {
    int r = idx >> 6, k4 = (idx & 63) << 2;
    size_t g = (size_t)(wg * 16 + r) * 256 + k4;
    float4 hv = *(const float4*)&h[g];
    float4 cv = *(const float4*)&c[g];
    hc[r * HCSTR + k4 + 0] = hv.x;
    hc[r * HCSTR + k4 + 1] = hv.y;
    hc[r * HCSTR + k4 + 2] = hv.z;
    hc[r * HCSTR + k4 + 3] = hv.w;
    hc[r * HCSTR + 256 + k4 + 0] = cv.x;
    hc[r * HCSTR + 256 + k4 + 1] = cv.y;
    hc[r * HCSTR + 256 + k4 + 2] = cv.z;
    hc[r * HCSTR + 256 + k4 + 3] = cv.w;
  }
  if (tid < 16) yt_s[tid] = ytl[wg * 16 + tid];
  __syncthreads();

  const int hl = lane >> 4, ml = lane & 15;
  v8f acc[4][2];                              // [gate i,f,g,o][2 N-subtiles]
#pragma unroll
  for (int g = 0; g < 4; ++g)
#pragma unroll
    for (int q = 0; q < 2; ++q) {
      int n = g * 256 + (wave * 2 + q) * 16 + ml;
      float bb = biasg[n], wv = Wih[n];
#pragma unroll
      for (int i = 0; i < 8; ++i) acc[g][q][i] = bb + yt_s[(hl << 3) + i] * wv;
    }
  for (int k0 = 0; k0 < 256; k0 += 4) {       // gates: K = DEC = 256
    int ka = k0 + (hl << 1);
    v2f a;
    a.x = hc[ml * HCSTR + ka];
    a.y = hc[ml * HCSTR + ka + 1];
#pragma unroll
    for (int g = 0; g < 4; ++g)
#pragma unroll
      for (int q = 0; q < 2; ++q) {
        int n = g * 256 + (wave * 2 + q) * 16 + ml;
        v2f bv = Whh2[(size_t)(ka >> 1) * 1024 + n];   // one b64 load
        acc[g][q] = wmma_f32_k4(a, bv, acc[g][q]);
      }
  }
  __syncthreads();                            // all A reads done before update

#pragma unroll
  for (int q = 0; q < 2; ++q)
#pragma unroll
    for (int i = 0; i < 8; ++i) {
      int r = (hl << 3) + i;
      int j = (wave * 2 + q) * 16 + ml;
      float iv = sigmoidf_(acc[0][q][i]);
      float fv = sigmoidf_(acc[1][q][i]);
      float gv = tanhf(acc[2][q][i]);
      float ov = sigmoidf_(acc[3][q][i]);
      float co = hc[r * HCSTR + 256 + j];
      float cn = fv * co + iv * gv;
      float hn = ov * tanhf(cn);
      hc[r * HCSTR + j]       = hn;
      hc[r * HCSTR + 256 + j] = cn;
      size_t g = (size_t)(wg * 16 + r) * 256 + j;
      h[g] = hn;
      c[g] = cn;
    }
  __syncthreads();

  v8f ua[2];                                  // u_next = [h,c] . W1hc^T (K=512)
#pragma unroll
  for (int q = 0; q < 2; ++q)
#pragma unroll
    for (int i = 0; i < 8; ++i) ua[q][i] = 0.0f;
  for (int k0 = 0; k0 < 512; k0 += 4) {
    int ka = k0 + (hl << 1);
    v2f a;
    a.x = hc[ml * HCSTR + ka];
    a.y = hc[ml * HCSTR + ka + 1];
#pragma unroll
    for (int q = 0; q < 2; ++q) {
      int n = (wave * 2 + q) * 16 + ml;
      v2f bv = Whc2[(size_t)(ka >> 1) * 256 + n];      // one b64 load
      ua[q] = wmma_f32_k4(a, bv, ua[q]);
    }
  }
#pragma unroll
  for (int q = 0; q < 2; ++q) {
    int n = (wave * 2 + q) * 16 + ml;
#pragma unroll
    for (int i = 0; i < 8; ++i) {
      int M = (hl << 3) + i;
      u[(size_t)(wg * 16 + M) * 256 + n] = ua[q][i];
    }
  }
}

// out[b] = [h[b], ctx[b]] . W_fcf^T + b_fcf
__global__ __launch_bounds__(256) void k_final(const float* __restrict__ h,
                                               const float* __restrict__ ctx,
                                               const float* __restrict__ Wfcf,
                                               const float* __restrict__ bfcf,
                                               float* __restrict__ out) {
  __shared__ float red[8];
  const int tid = threadIdx.x, b = blockIdx.x;
  const int lane = tid & 31, wave = tid >> 5;
  float v = h[(size_t)b * 256 + tid] * Wfcf[tid] +
            ctx[(size_t)b * 256 + tid] * Wfcf[256 + tid];
#pragma unroll
  for (int off = 16; off; off >>= 1) v += __shfl_xor(v, off, 32);
  if (lane == 0) red[wave] = v;
  __syncthreads();
  if (tid == 0) {
    float tot = bfcf[0];
#pragma unroll
    for (int w = 0; w < 8; ++w) tot += red[w];
    out[b] = tot;
  }
}

extern "C" void kernel_launch(void* const* d_in, const int* in_sizes, int n_in,
                              void* d_out, int out_size, void* d_ws, size_t ws_size,
                              hipStream_t stream) {
  (void)in_sizes; (void)n_in; (void)out_size; (void)ws_size;
  const float* enc   = (const float*)d_in[0];
  const float* yhist = (const float*)d_in[1];
  const float* W1    = (const float*)d_in[2];
  const float* b1    = (const float*)d_in[3];
  const float* W2    = (const float*)d_in[4];
  /* d_in[5] = b_attn2: softmax-invariant, unused */
  const float* Wih   = (const float*)d_in[6];
  const float* Whh   = (const float*)d_in[7];
  const float* bih   = (const float*)d_in[8];
  const float* bhh   = (const float*)d_in[9];
  const float* Wfc   = (const float*)d_in[10];
  const float* bfc   = (const float*)d_in[11];
  const float* Wfcf  = (const float*)d_in[12];
  const float* bfcf  = (const float*)d_in[13];

  float* out  = (float*)d_out;                // [512] then weight [512,127,127]
  float* wout = out + BB;

  float* ws = (float*)d_ws;
  size_t o = 0;
  float* Wt_hc = ws + o; o += 512 * 256;                 // W1 h|c cols, K-paired
  float* Wt_e  = ws + o; o += 256 * 256;                 // W1 enc cols, K-paired
  float* Wt_hh = ws + o; o += 256 * 1024;                // W_hh^T, K-paired
  float* biasg = ws + o; o += 1024;                      // b_ih + b_hh
  float* Eproj = ws + o; o += (size_t)BB * TM1 * 256;    // 66.6 MB (L2-resident)
  float* hbuf  = ws + o; o += BB * 256;
  float* cbuf  = ws + o; o += BB * 256;
  float* ubuf  = ws + o; o += BB * 256;                  // h.W1h^T + c.W1c^T
  float* ctxb  = ws + o; o += BB * 256;
  float* ytl   = ws + o; o += BB;

  k_prep_attn_w<<<768, 256, 0, stream>>>(W1, Wt_hc, Wt_e);
  k_prep_hh<<<1024, 256, 0, stream>>>(Whh, bih, bhh, Wt_hh, biasg);
  k_zero<<<(3 * BB * 256) / 256, 256, 0, stream>>>(hbuf, 3 * BB * 256); // h,c,u
  k_eproj<<<(BB * TM1) / 16, 256, 0, stream>>>(enc, Wt_e, b1, Eproj);

  for (int s = 0; s < TM1; ++s) {
    k_attn<<<BB, 256, 0, stream>>>(Eproj, enc, ubuf, W2, Wfc, bfc, yhist,
                                   ctxb, ytl, wout, s);
    k_lstm<<<BB / 16, 256, 0, stream>>>(hbuf, cbuf, ubuf, ytl, Wih, biasg,
                                        Wt_hh, Wt_hc);
  }
  k_final<<<BB, 256, 0, stream>>>(hbuf, ctxb, Wfcf, bfcf, out);
}